// Transfer_45320494907489
// MI455X (gfx1250) — compile-verified
//
#include <hip/hip_runtime.h>
#include <hip/hip_bf16.h>
#include <stdint.h>

// Problem constants (fixed by the reference)
#define B_  4
#define HW  96          // patch grid = lv2 spatial = 96x96
#define L_  (HW*HW)     // 9216
#define C2  128
#define C1  64
#define H1  192
#define CG  4           // channel groups per (b, m)
#define C2G (C2/CG)     // 32 lv2 channels per block
#define C1G (C1/CG)     // 16 lv1 channels per block

#if defined(__HIP_DEVICE_COMPILE__) && __has_builtin(__builtin_amdgcn_global_load_async_to_lds_b128)
#define USE_ASYNC_LDS 1
#else
#define USE_ASYNC_LDS 0
#endif

#if USE_ASYNC_LDS
typedef int v4i_ __attribute__((vector_size(16)));
typedef __attribute__((address_space(1))) v4i_* glb_v4i_ptr;
typedef __attribute__((address_space(3))) v4i_* lds_v4i_ptr;
#endif

__global__ __launch_bounds__(256)
void transfer_gather_fold_kernel(const int* __restrict__ idx,
                                 const float* __restrict__ ref1,   // [B,64,192,192]
                                 const float* __restrict__ ref2,   // [B,128,96,96]
                                 float* __restrict__ out2,         // [B,128,96,96]
                                 float* __restrict__ out1)         // [B,64,192,192]
{
    const int m   = blockIdx.x;   // patch row 0..95
    const int b   = blockIdx.y;   // batch
    const int g   = blockIdx.z;   // channel group
    const int tid = threadIdx.x;

    __shared__ __align__(16) int s_idx[3 * HW];   // 3 index rows (m-1, m, m+1)
    __shared__ int s_off2[HW * 9];                // lv2 plane offset per (pixel, tap), -1 = invalid
    __shared__ int s_off1[HW * 9];                // lv1 plane base offset per (pixel, tap)

    const int* idxb = idx + (size_t)b * L_;

    // ---- Phase 0: stage 3 index rows into LDS (CDNA5 async global->LDS path) ----
#if USE_ASYNC_LDS
    if (tid < 72) {                       // 3 rows * 96 ints = 72 x 16B transfers
        int jy = tid / 24, e = tid % 24;
        int py = m - 1 + jy;
        if (py >= 0 && py < HW) {
            const int* gsrc = idxb + py * HW + e * 4;
            int*       ldst = &s_idx[jy * HW + e * 4];
            __builtin_amdgcn_global_load_async_to_lds_b128(
                (glb_v4i_ptr)gsrc, (lds_v4i_ptr)ldst, 0, 0);
        }
    }
#if __has_builtin(__builtin_amdgcn_s_wait_asynccnt)
    __builtin_amdgcn_s_wait_asynccnt(0);
#else
    asm volatile("s_wait_asynccnt 0" ::: "memory");
#endif
#else
    for (int k = tid; k < 3 * HW; k += 256) {
        int jy = k / HW, e = k % HW;
        int py = m - 1 + jy;
        s_idx[k] = (py >= 0 && py < HW) ? idxb[py * HW + e] : 0;
    }
#endif
    __syncthreads();

    // ---- Phase 1: build the 9-tap offset tables for all 96 pixels of this row ----
    for (int k = tid; k < HW * 9; k += 256) {
        unsigned n = (unsigned)k / 9u;
        unsigned j = (unsigned)k % 9u;
        int jy = (int)(j / 3u), jx = (int)(j % 3u);
        int py = m + jy - 1;
        int px = (int)n + jx - 1;
        int off2 = -1, off1 = -1;
        if (py >= 0 && py < HW && px >= 0 && px < HW) {
            unsigned v  = (unsigned)s_idx[jy * HW + px];  // gathered patch index
            int ir = (int)(v / 96u);
            int ic = (int)(v % 96u);
            int r = ir - (jy - 1);          // lv2 source row  = m + (ir - py)
            int c = ic - (jx - 1);          // lv2 source col  = n + (ic - px)
            if (r >= 0 && r < HW && c >= 0 && c < HW) {  // same mask is valid for lv1
                off2 = r * HW + c;          // into 96x96 plane
                off1 = r * 384 + 2 * c;     // (2r)*192 + 2c into 192x192 plane
            }
        }
        s_off2[k] = off2;
        s_off1[k] = off1;
    }
    __syncthreads();

    const float inv9 = 1.0f / 9.0f;

    // ---- Phase 2a: T_lv2 — C2G channels x 96 pixels, coalesced NT stores ----
    for (int i = tid; i < C2G * HW; i += 256) {
        int n = i % HW;
        int c = g * C2G + i / HW;
        const float* plane = ref2 + (size_t)(b * C2 + c) * (size_t)L_;
        float s = 0.0f;
        #pragma unroll
        for (int j = 0; j < 9; ++j) {
            int o  = s_off2[n * 9 + j];
            int oc = (o < 0) ? 0 : o;          // clamp: load is always in-bounds
            float v = plane[oc];
            s += (o >= 0) ? v : 0.0f;
        }
        __builtin_nontemporal_store(s * inv9,
            &out2[((size_t)(b * C2 + c) * HW + m) * HW + n]);
    }

    // ---- Phase 2b: T_lv1 — C1G channels x 2 rows x 192 cols ----
    for (int i = tid; i < C1G * 2 * H1; i += 256) {
        int X  = i % H1;
        int t  = i / H1;
        int ys = t & 1;
        int c  = g * C1G + (t >> 1);
        int n  = X >> 1, xs = X & 1;
        int Y  = 2 * m + ys;
        int sub = ys * H1 + xs;
        const float* plane = ref1 + (size_t)(b * C1 + c) * (size_t)(H1 * H1);
        float s = 0.0f;
        #pragma unroll
        for (int j = 0; j < 9; ++j) {
            int o  = s_off1[n * 9 + j];
            int oc = (o < 0) ? 0 : o + sub;
            float v = plane[oc];
            s += (o >= 0) ? v : 0.0f;
        }
        __builtin_nontemporal_store(s * inv9,
            &out1[((size_t)(b * C1 + c) * H1 + Y) * H1 + X]);
    }
}

extern "C" void kernel_launch(void* const* d_in, const int* in_sizes, int n_in,
                              void* d_out, int out_size, void* d_ws, size_t ws_size,
                              hipStream_t stream) {
    // setup_inputs order: R_lv2_star_arg(int32), lrsr_lv2(f32, unused),
    //                     ref_lv1(f32), ref_lv2(f32)
    const int*   idx  = (const int*)  d_in[0];
    const float* ref1 = (const float*)d_in[2];
    const float* ref2 = (const float*)d_in[3];
    float* out2 = (float*)d_out;                              // [4,128,96,96]
    float* out1 = out2 + (size_t)B_ * C2 * HW * HW;           // [4,64,192,192]

    dim3 grid(HW, B_, CG);   // 96 x 4 x 4 = 1536 blocks
    dim3 block(256);         // 8 wave32
    transfer_gather_fold_kernel<<<grid, block, 0, stream>>>(idx, ref1, ref2, out2, out1);
}